// ConditionalAttention_24979529793593
// MI455X (gfx1250) — compile-verified
//
#include <hip/hip_runtime.h>
#include <hip/hip_bf16.h>

// ---------------------------------------------------------------------------
// Graph conditional attention for MI455X (gfx1250, wave32).
//  - All GEMMs via V_WMMA_F32_16X16X4_F32 (fp32-exact vs reference).
//  - `score` (327MB if materialized) never touches HBM: regs -> LDS -> WMMA.
//  - Weight matrices staged global->LDS once per block via the gfx1250
//    async-LDS path (GLOBAL_LOAD_ASYNC_TO_LDS_B128 / ASYNCcnt) when the
//    toolchain exposes the builtins; plain copies otherwise.
//  - Segmented softmax via order-preserving uint atomicMax + f32 atomicAdd.
// ---------------------------------------------------------------------------

#define N_NODES 40000
#define N_EDGES 640000
#define SCALE_F 0.25f          // HEAD_DIM^-0.5 = 16^-0.5

#define LDS_STRIDE 132         // score tile row stride (floats)
#define WSTRIDE    132         // weight tile row stride (floats), bank-conflict free

typedef float v2f __attribute__((ext_vector_type(2)));
typedef float v8f __attribute__((ext_vector_type(8)));
typedef int   v4i __attribute__((vector_size(16)));              // 16B chunk
typedef __attribute__((address_space(1))) v4i g_v4i;             // global
typedef __attribute__((address_space(3))) v4i l_v4i;             // LDS

#if __has_builtin(__builtin_amdgcn_global_load_async_to_lds_b128)
#define HAVE_ASYNC_LDS 1
#else
#define HAVE_ASYNC_LDS 0
#endif

__device__ __forceinline__ void wait_async0() {
#if HAVE_ASYNC_LDS
#if __has_builtin(__builtin_amdgcn_s_wait_asynccnt)
    __builtin_amdgcn_s_wait_asynccnt(0);
#else
    asm volatile("s_wait_asynccnt 0x0" ::: "memory");
#endif
#endif
}

__device__ __forceinline__ v8f wmma4(v2f a, v2f b, v8f c) {
    // D = A(16x4,f32) x B(4x16,f32) + C(16x16,f32)
    return __builtin_amdgcn_wmma_f32_16x16x4_f32(
        false, a, false, b, (short)0, c, false, false);
}

// Order-preserving float <-> uint encoding for atomicMax on floats.
__device__ __forceinline__ unsigned fenc(float f) {
    unsigned u = __float_as_uint(f);
    return (u & 0x80000000u) ? ~u : (u | 0x80000000u);
}
__device__ __forceinline__ float fdec(unsigned k) {
    unsigned u = (k & 0x80000000u) ? (k ^ 0x80000000u) : ~k;
    return __uint_as_float(u);
}

// Stage `rows` rows of a row-major (rows x 128) f32 weight slice into LDS with
// WSTRIDE padding. 128 threads cooperate; 16B chunks; async when available.
__device__ __forceinline__ void stage_weight(float* wlds, const float* __restrict__ W,
                                             int rows) {
    const int t = threadIdx.x;                 // 0..127
    const int nChunks = rows * 32;             // 16B chunks
    for (int idx = t; idx < nChunks; idx += 128) {
        const int row = idx >> 5;
        const int c4  = (idx & 31) * 4;
#if HAVE_ASYNC_LDS
        __builtin_amdgcn_global_load_async_to_lds_b128(
            (g_v4i*)(W + (size_t)row * 128 + c4),
            (l_v4i*)(wlds + row * WSTRIDE + c4),
            0, 0);
#else
        const float4 v = *(const float4*)(W + (size_t)row * 128 + c4);
        *(float4*)(wlds + row * WSTRIDE + c4) = v;
#endif
    }
}

// ---------------------------------------------------------------------------
// Generic fp32 WMMA GEMM: out(M x outDim) = A(M x 128) @ W(outDim x 128)^T + b
// One wave per 16x16 output tile; 4 waves (128 threads) per block.
// gridDim.x = M/16, gridDim.y = outDim/64. W slice for the block's 64 output
// columns is staged in LDS (async) and B-fragments are served by ds loads.
// ---------------------------------------------------------------------------
__global__ void gemm_nt_bias_kernel(const float* __restrict__ A,
                                    const float* __restrict__ W,
                                    const float* __restrict__ bias,
                                    float* __restrict__ out,
                                    int outDim) {
    __shared__ float wlds[64 * WSTRIDE];       // 33792 B

    const int lane = threadIdx.x & 31;
    const int wave = threadIdx.x >> 5;
    const int row0 = blockIdx.x * 16;
    const int col0 = (blockIdx.y * 4 + wave) * 16;

    stage_weight(wlds, W + (size_t)blockIdx.y * 64 * 128, 64);
    wait_async0();
    __syncthreads();

    const int mA    = lane & 15;          // A-frag row
    const int bk    = (lane >> 4) * 2;    // A/B-frag K sub-offset
    const int nB    = lane & 15;          // B-frag col
    const int mBase = (lane >> 4) * 8;    // C-frag row base

    v8f acc = {};
    const float* arow = A + (size_t)(row0 + mA) * 128 + bk;
    const float* brow = wlds + (wave * 16 + nB) * WSTRIDE + bk;
#pragma unroll
    for (int k = 0; k < 128; k += 4) {
        v2f a, b;
        a.x = arow[k]; a.y = arow[k + 1];
        b.x = brow[k]; b.y = brow[k + 1];
        acc = wmma4(a, b, acc);
    }
    const float bv = bias[col0 + (lane & 15)];
#pragma unroll
    for (int r = 0; r < 8; ++r) {
        out[(size_t)(row0 + mBase + r) * outDim + col0 + (lane & 15)] = acc[r] + bv;
    }
}

// ---------------------------------------------------------------------------
// Fused edge kernel: per wave, one 16-edge tile.
//   ctile  = tanh(e_tile @ c_w^T + c_b)          (WMMA, B-frags from LDS)
//   score  = k[src] * q[dst] * ctile             (gathers; qkv is L2-resident)
//   logit  = rowhead-sum(score)*SCALE -> ws, atomicMax into m[dst]
//   mean   = head-mean(score) -> d_out part 3    (overlapped w/ async restage)
//   e_out  = score @ eproj_w^T + eproj_b         (WMMA via LDS relayout)
// ---------------------------------------------------------------------------
__global__ void edge_fused_kernel(const float* __restrict__ e,
                                  const int* __restrict__ src,
                                  const int* __restrict__ dst,
                                  const float* __restrict__ c_w,
                                  const float* __restrict__ c_b,
                                  const float* __restrict__ eproj_w,
                                  const float* __restrict__ eproj_b,
                                  const float* __restrict__ qkv,
                                  float* __restrict__ logit,
                                  unsigned* __restrict__ menc,
                                  float* __restrict__ eout,
                                  float* __restrict__ meanout) {
    __shared__ float wlds[128 * WSTRIDE];      // 67584 B (c_w, then eproj_w)
    __shared__ float sc[4][16 * LDS_STRIDE];   // 33792 B

    const int lane = threadIdx.x & 31;
    const int wave = threadIdx.x >> 5;
    const int tile = blockIdx.x * 4 + wave;
    const int e0   = tile * 16;

    const int mA    = lane & 15;
    const int bk    = (lane >> 4) * 2;
    const int nB    = lane & 15;
    const int mBase = (lane >> 4) * 8;

    // Stage c_w (128x128) into LDS once per block.
    stage_weight(wlds, c_w, 128);
    wait_async0();
    __syncthreads();

    // ---- GEMM1: ctile = e_tile @ c_w^T (8 column tiles of 16) ----
    v8f acc[8];
#pragma unroll
    for (int t = 0; t < 8; ++t) acc[t] = {};
    {
        const float* arow = e + (size_t)(e0 + mA) * 128 + bk;
        for (int k = 0; k < 128; k += 4) {
            v2f a;
            a.x = arow[k]; a.y = arow[k + 1];
#pragma unroll
            for (int t = 0; t < 8; ++t) {
                const float* brow = wlds + (t * 16 + nB) * WSTRIDE + bk;
                v2f b;
                b.x = brow[k]; b.y = brow[k + 1];
                acc[t] = wmma4(a, b, acc[t]);
            }
        }
    }

    // ---- edge endpoints for this lane's 8 C-rows ----
    int srcs[8], dsts[8];
#pragma unroll
    for (int r = 0; r < 8; ++r) {
        srcs[r] = src[e0 + mBase + r];
        dsts[r] = dst[e0 + mBase + r];
    }

    // ---- score = k[src] * q[dst] * tanh(c + c_b); stage to LDS ----
#pragma unroll
    for (int t = 0; t < 8; ++t) {
        const int ch = t * 16 + (lane & 15);
        const float cb = c_b[ch];
#pragma unroll
        for (int r = 0; r < 8; ++r) {
            const float cv = tanhf(acc[t][r] + cb);
            const float kv = qkv[(size_t)srcs[r] * 384 + 128 + ch]; // k
            const float qv = qkv[(size_t)dsts[r] * 384 + ch];       // q
            sc[wave][(mBase + r) * LDS_STRIDE + ch] = kv * qv * cv;
        }
    }
    __syncthreads();   // sc visible; all GEMM1 LDS reads drained

    // Kick off async restage of eproj_w; overlap with reductions below.
    stage_weight(wlds, eproj_w, 128);

    // ---- logits: 16 rows x 8 heads, 4 tasks/lane ----
#pragma unroll
    for (int i = 0; i < 4; ++i) {
        const int task = lane + 32 * i;   // 0..127
        const int m  = task >> 3;
        const int hd = task & 7;
        float sum = 0.f;
#pragma unroll
        for (int j = 0; j < 16; ++j) sum += sc[wave][m * LDS_STRIDE + hd * 16 + j];
        const float lg = sum * SCALE_F;
        const int edge = e0 + m;
        logit[(size_t)edge * 8 + hd] = lg;
        atomicMax(&menc[(size_t)dst[edge] * 8 + hd], fenc(lg));
    }

    // ---- score.mean over heads: 16 rows x 16 cols, 8 tasks/lane ----
#pragma unroll
    for (int i = 0; i < 8; ++i) {
        const int task = lane + 32 * i;   // 0..255
        const int m = task >> 4;
        const int j = task & 15;
        float sum = 0.f;
#pragma unroll
        for (int hd = 0; hd < 8; ++hd) sum += sc[wave][m * LDS_STRIDE + hd * 16 + j];
        meanout[(size_t)(e0 + m) * 16 + j] = sum * 0.125f;
    }

    wait_async0();
    __syncthreads();   // eproj_w staged

    // ---- GEMM2: e_out = score @ eproj_w^T + eproj_b ----
#pragma unroll
    for (int t = 0; t < 8; ++t) acc[t] = {};
    {
        const float* lrow = &sc[wave][mA * LDS_STRIDE + bk];
        for (int k = 0; k < 128; k += 4) {
            v2f a;
            a.x = lrow[k]; a.y = lrow[k + 1];
#pragma unroll
            for (int t = 0; t < 8; ++t) {
                const float* brow = wlds + (t * 16 + nB) * WSTRIDE + bk;
                v2f b;
                b.x = brow[k]; b.y = brow[k + 1];
                acc[t] = wmma4(a, b, acc[t]);
            }
        }
    }
#pragma unroll
    for (int t = 0; t < 8; ++t) {
        const float bv = eproj_b[t * 16 + (lane & 15)];
#pragma unroll
        for (int r = 0; r < 8; ++r) {
            eout[(size_t)(e0 + mBase + r) * 128 + t * 16 + (lane & 15)] = acc[t][r] + bv;
        }
    }
}

// ---------------------------------------------------------------------------
// Pass 2 of softmax: ex = exp(logit - m[dst]); s[dst] += ex  (per edge,head)
// ---------------------------------------------------------------------------
__global__ void softmax_edge_kernel(float* __restrict__ exbuf,   // logit in, ex out
                                    const int* __restrict__ dst,
                                    const unsigned* __restrict__ menc,
                                    float* __restrict__ ssum) {
    const size_t tid = (size_t)blockIdx.x * 256 + threadIdx.x;  // E*8 exact
    const int edge = (int)(tid >> 3);
    const int hd   = (int)(tid & 7);
    const int d    = dst[edge];
    const float lg = exbuf[tid];
    const float mv = fdec(menc[(size_t)d * 8 + hd]);
    const float ex = __expf(lg - mv);
    exbuf[tid] = ex;
    atomicAdd(&ssum[(size_t)d * 8 + hd], ex);
}

// ---------------------------------------------------------------------------
// Message aggregation: h_agg[dst] += v[src] * (ex / s[dst])  (per edge,chan)
// ---------------------------------------------------------------------------
__global__ void msg_kernel(const float* __restrict__ exbuf,
                           const int* __restrict__ src,
                           const int* __restrict__ dst,
                           const float* __restrict__ ssum,
                           const float* __restrict__ qkv,
                           float* __restrict__ hagg) {
    const size_t tid = (size_t)blockIdx.x * 256 + threadIdx.x;  // E*128 exact
    const int edge = (int)(tid >> 7);
    const int ch   = (int)(tid & 127);
    const int hd   = ch >> 4;
    const int d = dst[edge];
    const int s = src[edge];
    const float attn = exbuf[(size_t)edge * 8 + hd] / ssum[(size_t)d * 8 + hd];
    const float vv   = qkv[(size_t)s * 384 + 256 + ch];          // v
    atomicAdd(&hagg[(size_t)d * 128 + ch], vv * attn);
}

// ---------------------------------------------------------------------------
// Scratch init: m = enc(-inf), s = 0, h_agg = 0
// ---------------------------------------------------------------------------
__global__ void init_kernel(unsigned* __restrict__ menc,
                            float* __restrict__ ssum,
                            float* __restrict__ hagg) {
    const int tid = blockIdx.x * 256 + threadIdx.x;
    if (tid < N_NODES * 8) {
        menc[tid] = 0x007FFFFFu;   // fenc(-inf)
        ssum[tid] = 0.f;
    }
    if (tid < N_NODES * 128) hagg[tid] = 0.f;
}

// ---------------------------------------------------------------------------
extern "C" void kernel_launch(void* const* d_in, const int* in_sizes, int n_in,
                              void* d_out, int out_size, void* d_ws, size_t ws_size,
                              hipStream_t stream) {
    const float* h       = (const float*)d_in[0];
    const float* e       = (const float*)d_in[1];
    const int*   src     = (const int*)d_in[2];
    const int*   dst     = (const int*)d_in[3];
    const float* qkv_w   = (const float*)d_in[4];
    const float* qkv_b   = (const float*)d_in[5];
    const float* c_w     = (const float*)d_in[6];
    const float* c_b     = (const float*)d_in[7];
    const float* hproj_w = (const float*)d_in[8];
    const float* hproj_b = (const float*)d_in[9];
    const float* eproj_w = (const float*)d_in[10];
    const float* eproj_b = (const float*)d_in[11];

    // Workspace carve-up (floats): ~105 MB total
    float*    qkv   = (float*)d_ws;                       // N*384
    float*    exbuf = qkv + (size_t)N_NODES * 384;        // E*8 (logit -> ex)
    unsigned* menc  = (unsigned*)(exbuf + (size_t)N_EDGES * 8);  // N*8
    float*    ssum  = (float*)(menc + (size_t)N_NODES * 8);      // N*8
    float*    hagg  = ssum + (size_t)N_NODES * 8;                // N*128

    // Output carve-up
    float* hout    = (float*)d_out;                        // N*128
    float* eout    = hout + (size_t)N_NODES * 128;         // E*128
    float* meanout = eout + (size_t)N_EDGES * 128;         // E*16

    // 1) init scratch
    init_kernel<<<(N_NODES * 128 + 255) / 256, 256, 0, stream>>>(menc, ssum, hagg);

    // 2) qkv = h @ qkv_w^T + qkv_b   (WMMA f32, weights via async LDS)
    gemm_nt_bias_kernel<<<dim3(N_NODES / 16, 384 / 64), 128, 0, stream>>>(
        h, qkv_w, qkv_b, qkv, 384);

    // 3) fused edge pipeline: c-gemm + score + logits + mean + e_out-gemm
    edge_fused_kernel<<<N_EDGES / 64, 128, 0, stream>>>(
        e, src, dst, c_w, c_b, eproj_w, eproj_b, qkv, exbuf, menc, eout, meanout);

    // 4) softmax pass 2
    softmax_edge_kernel<<<(N_EDGES * 8) / 256, 256, 0, stream>>>(exbuf, dst, menc, ssum);

    // 5) message scatter-add
    msg_kernel<<<((size_t)N_EDGES * 128) / 256, 256, 0, stream>>>(
        exbuf, src, dst, ssum, qkv, hagg);

    // 6) h_out = h_agg @ hproj_w^T + hproj_b   (WMMA f32, weights via async LDS)
    gemm_nt_bias_kernel<<<dim3(N_NODES / 16, 128 / 64), 128, 0, stream>>>(
        hagg, hproj_w, hproj_b, hout, 128);
}